// KANLayer_8675833938515
// MI455X (gfx1250) — compile-verified
//
#include <hip/hip_runtime.h>
#include <math.h>

// KAN layer forward for MI455X (gfx1250, wave32).
// Memory-bound (~810MB of output writes -> ~35us floor at 23.3 TB/s).
// Spline einsum (16b x 8k x 16o per input-dim) on V_WMMA_F32_16X16X4_F32
// (two K=4 steps); silu-base reduction out_base[b][o] = sum_i sb[i][o]*silu
// is also a 16x16xK GEMM on a third WMMA accumulator. De Boor divisions have
// wave-uniform denominators -> precomputed reciprocal table in LDS turns the
// recursion into pure mul/fma. Outputs staged through LDS so all big stores
// are contiguous b64 along the fastest (input-dim) axis.

#define BATCH   65536
#define IN_DIM  64
#define OUT_DIM 16
#define NB      8      // NUM + K spline coefficients
#define GRIDW   12     // NUM + 1 + 2K knots
#define KORD    3

#define WAVES   4      // waves per block, 16 batch rows each
#define ICHUNK  8      // input dims staged per flush

// reciprocal table: rd[p][t] = 1/(g[t+p]-g[t]); p=1:11, p=2:10, p=3:9 entries
#define RD_O1   0
#define RD_O2   11
#define RD_O3   21
#define RD_N    30
#define RD_PAD  32

typedef float f32x2 __attribute__((ext_vector_type(2)));
typedef float f32x8 __attribute__((ext_vector_type(8)));

__global__ __launch_bounds__(128) void kan_fwd_kernel(
    const float* __restrict__ x,          // [BATCH][IN_DIM]
    const float* __restrict__ grid,       // [IN_DIM][GRIDW]
    const float* __restrict__ coef,       // [IN_DIM][OUT_DIM][NB]
    const float* __restrict__ scale_base, // [IN_DIM][OUT_DIM]
    const float* __restrict__ scale_sp,   // [IN_DIM][OUT_DIM]
    float* __restrict__ out,              // [BATCH][OUT_DIM]
    float* __restrict__ preacts,          // [BATCH][OUT_DIM][IN_DIM]
    float* __restrict__ postacts,         // [BATCH][OUT_DIM][IN_DIM]
    float* __restrict__ postspline)       // [BATCH][OUT_DIM][IN_DIM]
{
    __shared__ float stage[WAVES][16][16][ICHUNK + 1]; // y-tile, i padded
    __shared__ float xbuf [WAVES][16][ICHUNK];
    __shared__ float sbuf [WAVES][16][ICHUNK];         // silu(x)
    __shared__ float rden [IN_DIM][RD_PAD];            // reciprocal knot diffs

    const int tid  = threadIdx.x;
    const int wave = tid >> 5;
    const int lane = tid & 31;
    const int hi   = lane >> 4;       // half-wave select (K / M split)
    const int low  = lane & 15;       // A: batch row;  B/C/D: out column
    const int b0   = (blockIdx.x * WAVES + wave) * 16;

    // ---- per-block precompute: reciprocals of (uniform) knot differences ----
    for (int e = tid; e < IN_DIM * RD_N; e += 128) {
        const int i = e / RD_N;
        const int r = e - i * RD_N;
        int p, t;
        if (r < RD_O2)      { p = 1; t = r; }
        else if (r < RD_O3) { p = 2; t = r - RD_O2; }
        else                { p = 3; t = r - RD_O3; }
        const float den = grid[i * GRIDW + t + p] - grid[i * GRIDW + t];
        rden[i][r] = 1.0f / den;
    }
    __syncthreads();

    float outSum[8];                  // spline part of out, C-layout
#pragma unroll
    for (int v = 0; v < 8; ++v) outSum[v] = 0.0f;
    f32x8 accOut = {};                // base part of out, WMMA accumulator

    for (int ic = 0; ic < IN_DIM / ICHUNK; ++ic) {
        for (int ii4 = 0; ii4 < ICHUNK; ii4 += 4) {
            f32x2 aS = {};            // silu A-fragment for the base GEMM
#pragma unroll
            for (int j = 0; j < 4; ++j) {
                const int ii = ii4 + j;
                const int i  = ic * ICHUNK + ii;

                // knot row for feature i (wave-uniform -> SMEM loads)
                float g[GRIDW];
#pragma unroll
                for (int t = 0; t < GRIDW; ++t) g[t] = grid[i * GRIDW + t];
                const float* rd = &rden[i][0];

                const float xv = x[(size_t)(b0 + low) * IN_DIM + i];

                // numerators reused across all recursion levels
                float xd[GRIDW];
#pragma unroll
                for (int t = 0; t < GRIDW; ++t) xd[t] = xv - g[t];

                // degree-0 indicator basis
                float B[GRIDW - 1];
#pragma unroll
                for (int t = 0; t < GRIDW - 1; ++t)
                    B[t] = (xv >= g[t] && xv < g[t + 1]) ? 1.0f : 0.0f;
                // De Boor recursion, divisions replaced by reciprocal table
#pragma unroll
                for (int p = 1; p <= KORD; ++p) {
                    const int off = (p == 1) ? RD_O1 : (p == 2) ? RD_O2 : RD_O3;
#pragma unroll
                    for (int t = 0; t < GRIDW - 1 - p; ++t) {
                        const float a = xd[t] * rd[off + t];
                        const float c = -xd[t + p + 1] * rd[off + t + 1];
                        B[t] = a * B[t] + c * B[t + 1];
                    }
                }
                // B[0..7]: basis values for (batch=b0+low, feature=i)

                const float sv = xv / (1.0f + __expf(-xv));   // silu
                if (hi == 0) {
                    xbuf[wave][low][ii] = xv;
                    sbuf[wave][low][ii] = sv;
                }
                // capture silu into the base-GEMM A fragment (K = v + 2*hi)
                aS.x = (j == 2 * hi)     ? sv : aS.x;
                aS.y = (j == 2 * hi + 1) ? sv : aS.y;

                // --- spline WMMA: y16x16 = A(16b x 8k) * coef_i(8k x 16o) ---
                // f32 16x16x4 layout: lane-half h, component v -> K = v + 2*h
                const float* cf = coef + (size_t)i * (OUT_DIM * NB) + low * NB;
                f32x2 a0, a1, bm0, bm1;
                a0.x  = hi ? B[2] : B[0];   a0.y  = hi ? B[3] : B[1];
                a1.x  = hi ? B[6] : B[4];   a1.y  = hi ? B[7] : B[5];
                bm0.x = cf[2 * hi];         bm0.y = cf[2 * hi + 1];
                bm1.x = cf[4 + 2 * hi];     bm1.y = cf[5 + 2 * hi];

                f32x8 acc = {};
                acc = __builtin_amdgcn_wmma_f32_16x16x4_f32(
                    false, a0, false, bm0, (short)0, acc, false, false);
                acc = __builtin_amdgcn_wmma_f32_16x16x4_f32(
                    false, a1, false, bm1, (short)0, acc, false, false);

                // epilogue: lane(low=o, hi), component v -> y[b0+v+8*hi][i][o]
                const float ss = scale_sp[i * OUT_DIM + low];
#pragma unroll
                for (int v = 0; v < 8; ++v) {
                    const float y = acc[v];
                    outSum[v] += ss * y;
                    stage[wave][v + 8 * hi][low][ii] = y;
                }
            }
            // --- base-term WMMA: accOut += silu(16b x 4i) * sb(4i x 16o) ---
            const int k0 = ic * ICHUNK + ii4 + 2 * hi;
            f32x2 bmS;
            bmS.x = scale_base[k0 * OUT_DIM + low];
            bmS.y = scale_base[(k0 + 1) * OUT_DIM + low];
            accOut = __builtin_amdgcn_wmma_f32_16x16x4_f32(
                false, aS, false, bmS, (short)0, accOut, false, false);
        }
        __syncthreads();

        // --- coalesced flush: i fastest, float2 per lane (32B per 4 lanes) ---
        const int ibase = ic * ICHUNK;
        for (int f = lane; f < 16 * 16 * (ICHUNK / 2); f += 32) {
            const int il = (f & 3) << 1;          // even i within chunk
            const int o  = (f >> 2) & 15;
            const int bl = f >> 6;
            const int i  = ibase + il;
            f32x2 y, xv, base, sb, ss, pa;
            y.x    = stage[wave][bl][o][il];      y.y    = stage[wave][bl][o][il + 1];
            xv.x   = xbuf[wave][bl][il];          xv.y   = xbuf[wave][bl][il + 1];
            base.x = sbuf[wave][bl][il];          base.y = sbuf[wave][bl][il + 1];
            sb.x   = scale_base[i * OUT_DIM + o]; sb.y   = scale_base[(i + 1) * OUT_DIM + o];
            ss.x   = scale_sp  [i * OUT_DIM + o]; ss.y   = scale_sp  [(i + 1) * OUT_DIM + o];
            pa.x = sb.x * base.x + ss.x * y.x;
            pa.y = sb.y * base.y + ss.y * y.y;
            const size_t idx = ((size_t)(b0 + bl) * OUT_DIM + o) * IN_DIM + i;
            *(f32x2*)(postspline + idx) = y;
            *(f32x2*)(postacts   + idx) = pa;
            *(f32x2*)(preacts    + idx) = xv;
        }
        __syncthreads();   // stage reused next chunk
    }

    // out[b][o] = base GEMM + spline sum; each (b,o) owned by one (lane,v)
#pragma unroll
    for (int v = 0; v < 8; ++v)
        out[(size_t)(b0 + v + 8 * hi) * OUT_DIM + low] = accOut[v] + outSum[v];
}

extern "C" void kernel_launch(void* const* d_in, const int* in_sizes, int n_in,
                              void* d_out, int out_size, void* d_ws, size_t ws_size,
                              hipStream_t stream) {
    const float* x          = (const float*)d_in[0];
    const float* grid       = (const float*)d_in[1];
    const float* coef       = (const float*)d_in[2];
    const float* scale_base = (const float*)d_in[3];
    const float* scale_sp   = (const float*)d_in[4];

    float* out        = (float*)d_out;                       // [BATCH][OUT_DIM]
    float* preacts    = out + (size_t)BATCH * OUT_DIM;       // [BATCH][OUT_DIM][IN_DIM]
    float* postacts   = preacts  + (size_t)BATCH * OUT_DIM * IN_DIM;
    float* postspline = postacts + (size_t)BATCH * OUT_DIM * IN_DIM;

    const int blocks = BATCH / (WAVES * 16);   // 1024
    kan_fwd_kernel<<<blocks, WAVES * 32, 0, stream>>>(
        x, grid, coef, scale_base, scale_sp,
        out, preacts, postacts, postspline);
}